// DifferentiableSorter_63857573757715
// MI455X (gfx1250) — compile-verified
//
#include <hip/hip_runtime.h>
#include <math.h>

typedef __attribute__((ext_vector_type(2))) float v2f;
typedef __attribute__((ext_vector_type(4))) float v4f;
typedef __attribute__((ext_vector_type(8))) float v8f;

#define NN 4096
#define DD 64
#define NITERS 50

// Online logsumexp update with a SINGLE exp per element:
// one of exp(mx-nm)/exp(x-nm) is always exp(0)=1, so compute e=exp(-|x-mx|)
// and select which side it scales. Branchless (v_cndmask), 1 trans op.
__device__ __forceinline__ void lse_update(float& mx, float& sm, float x) {
  const float e = __expf(-fabsf(x - mx));
  sm = (x > mx) ? __fmaf_rn(sm, e, 1.0f) : (sm + e);
  mx = fmaxf(mx, x);
}
__device__ __forceinline__ void lse_combine(float& mx, float& sm,
                                            float omx, float osm) {
  const float e = __expf(-fabsf(omx - mx));
  sm = (omx > mx) ? __fmaf_rn(sm, e, osm) : __fmaf_rn(osm, e, sm);
  mx = fmaxf(mx, omx);
}

// ---------------------------------------------------------------------------
// GEMM: A[n][m] = sum_d X0[n][d] * W[m][d] + b[m]
// via V_WMMA_F32_16X16X4_F32, one 16x16 tile per wave, K=64 fully unrolled.
// ---------------------------------------------------------------------------
__global__ __launch_bounds__(128) void sinkhorn_gemm_kernel(
    const float* __restrict__ X, const float* __restrict__ W,
    const float* __restrict__ bias, float* __restrict__ A) {
  const int wave = threadIdx.x >> 5;
  const int lane = threadIdx.x & 31;
  const int half = lane >> 4;
  const int lm   = lane & 15;

  const int tileRow = blockIdx.x >> 6;                   // 0..255
  const int tileCol = ((blockIdx.x & 63) << 2) + wave;   // 0..255
  const int n_base = tileRow << 4;
  const int m_base = tileCol << 4;

  const float* aPtr = X + (n_base + lm) * DD + 2 * half; // batch 0 of X
  const float* bPtr = W + (m_base + lm) * DD + 2 * half;

  v8f acc = {};
#pragma unroll
  for (int ks = 0; ks < 16; ++ks) {                      // K = 64 = 16 * 4
    v2f afrag = *(const v2f*)(aPtr + 4 * ks);
    v2f bfrag = *(const v2f*)(bPtr + 4 * ks);
    acc = __builtin_amdgcn_wmma_f32_16x16x4_f32(false, afrag, false, bfrag,
                                                (short)0, acc, false, false);
  }

  const float bv = bias[m_base + lm];
#pragma unroll
  for (int i = 0; i < 8; ++i) {
    // VGPR i holds row M=i (lanes 0-15) and M=i+8 (lanes 16-31)
    A[(n_base + i + 8 * half) * NN + m_base + lm] = acc[i] + bv;
  }
}

__global__ void sinkhorn_zero_kernel(float* __restrict__ u) {
  u[blockIdx.x * 256 + threadIdx.x] = 0.0f;
}

// ---------------------------------------------------------------------------
// Column pass: v[m] = -logsumexp_n(A[n,m] + u[n]).
// Block = 256 threads covers 32 columns. Each thread owns a float4 of
// columns (c4 = tid&7) and one of 32 row-groups (rg = tid>>3); b128 loads
// move 512B per wave instruction. 32 partials/column combined via LDS tree.
// ---------------------------------------------------------------------------
__global__ __launch_bounds__(256) void sinkhorn_colpass_kernel(
    const float* __restrict__ A, const float* __restrict__ u,
    float* __restrict__ v) {
  __shared__ v4f smx[32][8];
  __shared__ v4f ssm[32][8];
  const int c4 = threadIdx.x & 7;        // which float4 of the 32 columns
  const int rg = threadIdx.x >> 3;       // row group 0..31
  const int m4 = (blockIdx.x << 3) + c4; // float4 column index
  const v4f* A4 = (const v4f*)A;

  float mx4[4] = {-INFINITY, -INFINITY, -INFINITY, -INFINITY};
  float sm4[4] = {0.0f, 0.0f, 0.0f, 0.0f};
  for (int n = rg; n < NN; n += 32) {
    v4f a4 = A4[n * (NN / 4) + m4];
    const float uv = u[n];
#pragma unroll
    for (int j = 0; j < 4; ++j) lse_update(mx4[j], sm4[j], a4[j] + uv);
  }
  {
    v4f tm, ts;
#pragma unroll
    for (int j = 0; j < 4; ++j) { tm[j] = mx4[j]; ts[j] = sm4[j]; }
    smx[rg][c4] = tm;
    ssm[rg][c4] = ts;
  }
  __syncthreads();
#pragma unroll
  for (int h = 16; h >= 1; h >>= 1) {
    if (rg < h) {
      v4f omx = smx[rg + h][c4], osm = ssm[rg + h][c4];
#pragma unroll
      for (int j = 0; j < 4; ++j) lse_combine(mx4[j], sm4[j], omx[j], osm[j]);
      v4f tm, ts;
#pragma unroll
      for (int j = 0; j < 4; ++j) { tm[j] = mx4[j]; ts[j] = sm4[j]; }
      smx[rg][c4] = tm;
      ssm[rg][c4] = ts;
    }
    __syncthreads();
  }
  if (rg == 0) {
    v4f out;
#pragma unroll
    for (int j = 0; j < 4; ++j) out[j] = -(mx4[j] + __logf(sm4[j]));
    ((v4f*)v)[m4] = out;
  }
}

// ---------------------------------------------------------------------------
// Row pass: u[n] = -logsumexp_m(A[n,m] + v[m]).
// One wave32 per row; each lane streams float4 chunks (512B/wave/instr),
// then shfl_xor tree combine of the online-LSE state.
// ---------------------------------------------------------------------------
__global__ __launch_bounds__(256) void sinkhorn_rowpass_kernel(
    const float* __restrict__ A, const float* __restrict__ v,
    float* __restrict__ u) {
  const int lane = threadIdx.x & 31;
  const int wave = threadIdx.x >> 5;
  const int n    = (blockIdx.x << 3) + wave;

  const v4f* A4row = (const v4f*)(A + n * NN);
  const v4f* V4    = (const v4f*)v;

  float mx = -INFINITY, sm = 0.0f;
#pragma unroll 4
  for (int i = 0; i < NN / 4 / 32; ++i) {
    const int idx = i * 32 + lane;
    v4f a4 = A4row[idx];
    v4f v4 = V4[idx];
#pragma unroll
    for (int j = 0; j < 4; ++j) lse_update(mx, sm, a4[j] + v4[j]);
  }
#pragma unroll
  for (int mask = 16; mask >= 1; mask >>= 1) {
    float omx = __shfl_xor(mx, mask, 32);
    float osm = __shfl_xor(sm, mask, 32);
    lse_combine(mx, sm, omx, osm);
  }
  if (lane == 0) u[n] = -(mx + __logf(sm));
}

// ---------------------------------------------------------------------------
// Finalize: out[n,m] = exp(A[n,m] + u[n] + v[m]), in place, float4 vectorized.
// ---------------------------------------------------------------------------
__global__ __launch_bounds__(256) void sinkhorn_finalize_kernel(
    float* __restrict__ A, const float* __restrict__ u,
    const float* __restrict__ v) {
  const int idx4 = blockIdx.x * 256 + threadIdx.x;  // index in float4 units
  const int n    = idx4 >> 10;                      // 1024 float4 per row
  const int m4   = idx4 & 1023;
  v4f a  = ((v4f*)A)[idx4];
  v4f vv = ((const v4f*)v)[m4];
  const float un = u[n];
#pragma unroll
  for (int i = 0; i < 4; ++i) a[i] = __expf(a[i] + un + vv[i]);
  ((v4f*)A)[idx4] = a;
}

extern "C" void kernel_launch(void* const* d_in, const int* in_sizes, int n_in,
                              void* d_out, int out_size, void* d_ws, size_t ws_size,
                              hipStream_t stream) {
  const float* X = (const float*)d_in[0];  // (4, 4096, 64) - only batch 0 used
  const float* W = (const float*)d_in[1];  // (4096, 64)
  const float* b = (const float*)d_in[2];  // (4096,)

  float* A = (float*)d_out;                // 4096x4096 log_alpha, becomes output
  float* u = (float*)d_ws;                 // 4096 row potentials
  float* v = u + NN;                       // 4096 col potentials (16B aligned)

  // log_alpha for batch 0 (WMMA f32 GEMM + bias), staged in d_out
  sinkhorn_gemm_kernel<<<16384, 128, 0, stream>>>(X, W, b, A);
  sinkhorn_zero_kernel<<<NN / 256, 256, 0, stream>>>(u);

  // 50 Sinkhorn iterations reduced to potential-vector updates over fixed A
  // (A = 64MB stays resident in the 192MB L2 across all 100 passes)
  for (int it = 0; it < NITERS; ++it) {
    sinkhorn_colpass_kernel<<<NN / 32, 256, 0, stream>>>(A, u, v);
    sinkhorn_rowpass_kernel<<<NN / 8, 256, 0, stream>>>(A, v, u);
  }

  // out = exp(A + u + v)
  sinkhorn_finalize_kernel<<<(NN * (NN / 4)) / 256, 256, 0, stream>>>(A, u, v);
}